// Conv1dfft_19043884990783
// MI455X (gfx1250) — compile-verified
//
#include <hip/hip_runtime.h>
#include <math.h>

// ---------------- problem constants (from the reference) ----------------
#define N_B    32
#define C_B    64
#define W_LEN  4096
#define F_B    128
#define WW_B   9
#define S_FFT  8215      // fft_size = W + 2*PAD + 2*(WW-1) + (out_W-1)
#define HALF   2053      // spectrum bins kept after INDEX_BACK=50 truncation
#define K16    2064      // HALF rounded up to a multiple of 16 (WMMA tiling)
#define KTOT   4128      // 2*K16: [Re | -Im] concatenated contraction
#define NC     2048      // N*C
#define NF     4096      // N*F
#define OUTW   4096
#define LEFTX  12        // filter_pad + PAD: x's shift inside the padded signal
#define TWO_PI 6.283185307179586476925f

typedef float v2f __attribute__((ext_vector_type(2)));
typedef float v8f __attribute__((ext_vector_type(8)));
typedef unsigned int u32x4 __attribute__((ext_vector_type(4)));
typedef int i32x8 __attribute__((ext_vector_type(8)));
typedef int i32x4 __attribute__((ext_vector_type(4)));

// D = A(16x4, f32) * B(4x16, f32) + C(16x16, f32)  -- CDNA5 V_WMMA_F32_16X16X4_F32
__device__ __forceinline__ v8f wmma_f32(v2f a, v2f b, v8f c) {
  return __builtin_amdgcn_wmma_f32_16x16x4_f32(false, a, false, b, (short)0, c,
                                               false, false);
}

// ---------------------------------------------------------------------------
// Kernel 1: forward DFT twiddles (k-major, feeds the A fragments).
//   Cmat[k][t] =  cos(2*pi*k*(t+12)/S)   (k < HALF, else 0)
//   Smat[k][t] = -sin(2*pi*k*(t+12)/S)   so Xre = Cmat*x, Xim = Smat*x
// Exact integer phase reduction: k*(t+12) <= 2052*4107 ~ 8.4e6 < 2^24.
// ---------------------------------------------------------------------------
__global__ void __launch_bounds__(256)
gen_fwd_twiddle_kernel(float* __restrict__ Cmat, float* __restrict__ Smat) {
  int i = blockIdx.x * blockDim.x + threadIdx.x;
  if (i >= K16 * W_LEN) return;
  int k = i >> 12;      // / 4096
  int t = i & 4095;
  float cv = 0.f, sv = 0.f;
  if (k < HALF) {
    int ph = (int)(((long long)k * (long long)(t + LEFTX)) % S_FFT);
    float th = (float)ph * (TWO_PI / (float)S_FFT);
    float s, c;
    __sincosf(th, &s, &c);
    cv = c;
    sv = -s;
  }
  Cmat[i] = cv;
  Smat[i] = sv;
}

// ---------------------------------------------------------------------------
// Kernel 2: forward DFT as GEMM. One wave -> 16(k) x 64(nc) tile:
// per K-step (t += 4): 2 A-fragment v2f loads (cos,sin) + 4 B-fragment v2f
// loads, 8 WMMAs. All fragments are contiguous float2 per lane:
//   A lane(m,h): M=m, K=2h..2h+1 -> Cmat[k0+m][t+2h..]
//   B lane(m,h): N=m, K=2h..2h+1 -> x[nc+m][t+2h..]
// ---------------------------------------------------------------------------
__global__ void __launch_bounds__(128)
fwd_dft_gemm_kernel(const float* __restrict__ x,
                    const float* __restrict__ Cmat,
                    const float* __restrict__ Smat,
                    float* __restrict__ Xre,
                    float* __restrict__ Xim) {
  const int lane = threadIdx.x & 31;
  const int wv   = threadIdx.x >> 5;
  const int m = lane & 15;
  const int h = lane >> 4;
  const int k0  = blockIdx.x * 16;
  const int nc0 = (blockIdx.y * 4 + wv) * 64;

  const float* ac_row = Cmat + (size_t)(k0 + m) * W_LEN + 2 * h;
  const float* as_row = Smat + (size_t)(k0 + m) * W_LEN + 2 * h;
  const float* b_row  = x    + (size_t)(nc0 + m) * W_LEN + 2 * h;

  v8f z = {};
  v8f are[4] = {z, z, z, z};
  v8f aim[4] = {z, z, z, z};
#pragma unroll 2
  for (int t = 0; t < W_LEN; t += 4) {
    v2f ac = *(const v2f*)(ac_row + t);
    v2f as = *(const v2f*)(as_row + t);
#pragma unroll
    for (int j = 0; j < 4; ++j) {
      v2f bb = *(const v2f*)(b_row + (size_t)j * 16 * W_LEN + t);
      are[j] = wmma_f32(ac, bb, are[j]);
      aim[j] = wmma_f32(as, bb, aim[j]);
    }
  }
#pragma unroll
  for (int j = 0; j < 4; ++j) {
#pragma unroll
    for (int i = 0; i < 8; ++i) {              // D: row i+8h, col m
      int kk = k0 + i + 8 * h;
      int nc = nc0 + j * 16 + m;
      Xre[(size_t)kk * NC + nc] = are[j][i];
      Xim[(size_t)kk * NC + nc] = aim[j][i];
    }
  }
}

// ---------------------------------------------------------------------------
// Kernel 3: per-bin spectral multiply (einsum nck,fck->nfk with conj(W)).
// One block per frequency bin k. The X slice (Re row + Im row, adjacent
// workspace buffers) is staged into LDS by the Tensor Data Mover as one 2D
// tile: tile 2048x2, row stride K16*NC elements. Wave 0 builds the D#
// (ISA ch. 8 layout), issues tensor_load_to_lds (6-arg toolchain form),
// waits TENSORcnt==0, then the workgroup barrier publishes the data. w's
// 9-tap spectrum is built on the fly with an incremental complex rotation.
// Writes Oft[r][k] = Re, Oft[r][K16+k] = -Im so the inverse transform is one
// real GEMM with K = 2*K16.
// ---------------------------------------------------------------------------
__global__ void __launch_bounds__(256)
spectral_mul_kernel(const float* __restrict__ w,
                    const float* __restrict__ Xre,   // Xim = Xre + K16*NC
                    float* __restrict__ Oft) {
  __shared__ float sX[2 * NC];       // [0,2048) = Re, [2048,4096) = Im
  __shared__ float sWre[64 * C_B];   // half of F at a time
  __shared__ float sWim[64 * C_B];

  const int k   = blockIdx.x;
  const int tid = threadIdx.x;

  if (k >= HALF) {                   // zero the padded contraction columns
    for (int r = tid; r < NF; r += 256) {
      Oft[(size_t)r * KTOT + k]       = 0.f;
      Oft[(size_t)r * KTOT + K16 + k] = 0.f;
    }
    return;
  }

  if (tid < 32) {                    // wave 0 drives the TDM
    unsigned long long ga =
        (unsigned long long)(uintptr_t)(Xre + (size_t)k * NC);
    unsigned int lds = (unsigned int)(uintptr_t)(&sX[0]);  // LDS byte offset
    u32x4 g0;
    g0.x = 1u;                                   // count=1 (valid user D#)
    g0.y = lds;                                  // lds_addr
    g0.z = (unsigned int)ga;                     // global_addr[31:0]
    g0.w = (unsigned int)((ga >> 32) & 0x01FFFFFFu) | (2u << 30);  // +type=2
    i32x8 g1;
    g1[0] = (int)(2u << 16);                     // data_size = 4 bytes
    g1[1] = (int)((NC & 0xFFFFu) << 16);         // tensor_dim0[15:0]
    g1[2] = (int)(((NC >> 16) & 0xFFFFu) | (2u << 16));  // dim0 hi | dim1 = 2
    g1[3] = (int)((NC & 0xFFFFu) << 16);         // tile_dim0 = 2048
    g1[4] = (int)2;                              // tile_dim1 = 2
    g1[5] = (int)((unsigned)K16 * (unsigned)NC); // tensor_dim0_stride[31:0]
    g1[6] = 0;                                   // stride hi / dim1_stride
    g1[7] = 0;
    i32x4 zz4 = {0, 0, 0, 0};
    i32x8 zz8 = {0, 0, 0, 0, 0, 0, 0, 0};
    __builtin_amdgcn_tensor_load_to_lds(g0, g1, zz4, zz4, zz8, 0);
    __builtin_amdgcn_s_wait_tensorcnt(0);
  }
  __syncthreads();

  const float* sXre = sX;
  const float* sXim = sX + NC;

  const float thd = (float)k * (TWO_PI / (float)S_FFT);   // k < S, exact
  float sd, cd;
  __sincosf(thd, &sd, &cd);
  const int   p8  = (8 * k) % S_FFT;
  const float th8 = (float)p8 * (TWO_PI / (float)S_FFT);

  for (int fh = 0; fh < 2; ++fh) {
    __syncthreads();
    // conj(W)[f][c] = sum_u w[f][c][u] * (cos((u+8)d) + i*sin((u+8)d))
    for (int i = tid; i < 64 * C_B; i += 256) {
      const float* wp = w + (size_t)(fh * 64 * C_B + i) * WW_B;
      float cr, sr;
      __sincosf(th8, &sr, &cr);
      float are = 0.f, aim = 0.f;
#pragma unroll
      for (int u = 0; u < WW_B; ++u) {
        float wval = wp[u];
        are += wval * cr;
        aim += wval * sr;
        float c2 = cr * cd - sr * sd;              // rotate by d
        sr = sr * cd + cr * sd;
        cr = c2;
      }
      sWre[i] = are;
      sWim[i] = aim;
    }
    __syncthreads();
    // O[n][f] = sum_c X[n][c] * conjW[f][c]   (complex)
    for (int o = tid; o < N_B * 64; o += 256) {
      int n  = o >> 6;
      int fl = o & 63;
      int f  = fh * 64 + fl;
      const float* xr = sXre + n * C_B;
      const float* xi = sXim + n * C_B;
      const float* wr = sWre + fl * C_B;
      const float* wi = sWim + fl * C_B;
      float ore = 0.f, oim = 0.f;
#pragma unroll 8
      for (int c = 0; c < C_B; ++c) {
        float a = xr[c], bb = xi[c], p = wr[c], q = wi[c];
        ore += a * p - bb * q;
        oim += a * q + bb * p;
      }
      size_t r = (size_t)(n * F_B + f);
      Oft[r * KTOT + k]       = ore;
      Oft[r * KTOT + K16 + k] = -oim;
    }
  }
}

// ---------------------------------------------------------------------------
// Kernel 4: inverse DFT twiddles, stored TRANSPOSED (t-major) so the inverse
// GEMM's B fragments are contiguous float2 loads. Overwrites Cmat/Smat
// (stream order serializes this after the forward GEMM).
//   ICcT[t][k] = scale_k * cos(2*pi*k*t/S), ICsT[t][k] = scale_k * sin(...)
//   scale_0 = 1/S, scale_k = 2/S (S odd -> no Nyquist bin among kept k).
// ---------------------------------------------------------------------------
__global__ void __launch_bounds__(256)
gen_inv_twiddle_kernel(float* __restrict__ ICcT, float* __restrict__ ICsT) {
  const int t = blockIdx.x;
  for (int k = threadIdx.x; k < K16; k += 256) {
    float cv = 0.f, sv = 0.f;
    if (k < HALF) {
      int ph = (int)(((long long)k * (long long)t) % S_FFT);
      float th = (float)ph * (TWO_PI / (float)S_FFT);
      float s, c;
      __sincosf(th, &s, &c);
      float scale = (k == 0) ? (1.0f / (float)S_FFT) : (2.0f / (float)S_FFT);
      cv = scale * c;
      sv = scale * s;
    }
    ICcT[(size_t)t * K16 + k] = cv;
    ICsT[(size_t)t * K16 + k] = sv;
  }
}

// ---------------------------------------------------------------------------
// Kernel 5: inverse DFT as GEMM + fused bias. One wave -> 32(r) x 64(t):
// per K-step: 2 A v2f loads + 4 B v2f loads, 8 WMMAs. K loop split into the
// Re half (cos) then the -Im half (sin).
//   out[r][t] = sum_k Ore[r][k]*ICc[k][t] + (-Oim[r][k])*ICs[k][t] + b[r&127]
// ---------------------------------------------------------------------------
__global__ void __launch_bounds__(128)
inv_dft_gemm_bias_kernel(const float* __restrict__ Oft,
                         const float* __restrict__ ICcT,
                         const float* __restrict__ ICsT,
                         const float* __restrict__ bias,
                         float* __restrict__ out) {
  const int lane = threadIdx.x & 31;
  const int wv   = threadIdx.x >> 5;
  const int m = lane & 15;
  const int h = lane >> 4;
  const int r0 = blockIdx.x * 32;
  const int t0 = (blockIdx.y * 4 + wv) * 64;

  const float* a0 = Oft + (size_t)(r0 + m) * KTOT + 2 * h;
  const float* a1 = a0 + (size_t)16 * KTOT;
  const float* bc = ICcT + (size_t)(t0 + m) * K16 + 2 * h;
  const float* bs = ICsT + (size_t)(t0 + m) * K16 + 2 * h;

  v8f z = {};
  v8f acc[2][4] = {{z, z, z, z}, {z, z, z, z}};
#pragma unroll 2
  for (int k = 0; k < K16; k += 4) {           // Re(O) * cos
    v2f av0 = *(const v2f*)(a0 + k);
    v2f av1 = *(const v2f*)(a1 + k);
#pragma unroll
    for (int j = 0; j < 4; ++j) {
      v2f bb = *(const v2f*)(bc + (size_t)j * 16 * K16 + k);
      acc[0][j] = wmma_f32(av0, bb, acc[0][j]);
      acc[1][j] = wmma_f32(av1, bb, acc[1][j]);
    }
  }
#pragma unroll 2
  for (int k = 0; k < K16; k += 4) {           // (-Im(O)) * sin
    v2f av0 = *(const v2f*)(a0 + K16 + k);
    v2f av1 = *(const v2f*)(a1 + K16 + k);
#pragma unroll
    for (int j = 0; j < 4; ++j) {
      v2f bb = *(const v2f*)(bs + (size_t)j * 16 * K16 + k);
      acc[0][j] = wmma_f32(av0, bb, acc[0][j]);
      acc[1][j] = wmma_f32(av1, bb, acc[1][j]);
    }
  }
#pragma unroll
  for (int i2 = 0; i2 < 2; ++i2) {
#pragma unroll
    for (int j = 0; j < 4; ++j) {
#pragma unroll
      for (int i = 0; i < 8; ++i) {
        int r = r0 + i2 * 16 + i + 8 * h;
        out[(size_t)r * OUTW + t0 + j * 16 + m] = acc[i2][j][i] + bias[r & 127];
      }
    }
  }
}

// ---------------------------------------------------------------------------
// Workspace layout (floats):
//   Cmat : K16*W_LEN = 8,454,144   (fwd cos; later overwritten with ICcT)
//   Smat : 8,454,144               (fwd -sin; later overwritten with ICsT)
//   Xre  : K16*NC    = 4,227,072   (Xim must directly follow: TDM 2D tile)
//   Xim  : 4,227,072
//   Oft  : NF*KTOT   = 16,908,288
// total 42,270,720 floats = 169,082,880 bytes (~169 MB)
// ---------------------------------------------------------------------------
extern "C" void kernel_launch(void* const* d_in, const int* in_sizes, int n_in,
                              void* d_out, int out_size, void* d_ws,
                              size_t ws_size, hipStream_t stream) {
  (void)in_sizes; (void)n_in; (void)out_size; (void)ws_size;
  const float* x = (const float*)d_in[0];   // (32, 64, 4096)
  const float* w = (const float*)d_in[1];   // (128, 64, 9)
  const float* b = (const float*)d_in[2];   // (128,)
  float* out = (float*)d_out;               // (32, 128, 4096)

  float* wsf  = (float*)d_ws;
  float* Cmat = wsf;
  float* Smat = Cmat + (size_t)K16 * W_LEN;
  float* Xre  = Smat + (size_t)K16 * W_LEN;
  float* Xim  = Xre + (size_t)K16 * NC;     // contiguous after Xre (TDM tile)
  float* Oft  = Xim + (size_t)K16 * NC;

  const int TWID_GRID = (K16 * W_LEN + 255) / 256;   // 33024

  gen_fwd_twiddle_kernel<<<TWID_GRID, 256, 0, stream>>>(Cmat, Smat);

  // 129 k-tiles x (8 blocks * 4 waves -> 32 nc-supertiles of 64)
  fwd_dft_gemm_kernel<<<dim3(129, 8), 128, 0, stream>>>(x, Cmat, Smat, Xre,
                                                        Xim);

  spectral_mul_kernel<<<K16, 256, 0, stream>>>(w, Xre, Oft);

  gen_inv_twiddle_kernel<<<W_LEN, 256, 0, stream>>>(Cmat, Smat);

  // 128 r-supertiles of 32 x (16 blocks * 4 waves -> 64 t-supertiles of 64)
  inv_dft_gemm_bias_kernel<<<dim3(128, 16), 128, 0, stream>>>(Oft, Cmat, Smat,
                                                              b, out);
}